// CEMR_81346680586755
// MI455X (gfx1250) — compile-verified
//
#include <hip/hip_runtime.h>
#include <hip/hip_bf16.h>
#include <math.h>
#include <stdint.h>

typedef float v2f __attribute__((ext_vector_type(2)));
typedef float v8f __attribute__((ext_vector_type(8)));

#define ACT_NONE    0
#define ACT_RELU    1
#define ACT_SILU    2
#define ACT_SIGMOID 3

__device__ __forceinline__ float act_apply(float v, int act) {
    if (act == ACT_RELU)    return v > 0.f ? v : 0.f;
    if (act == ACT_SILU)    return v / (1.f + __expf(-v));      // x*sigmoid(x)
    if (act == ACT_SIGMOID) return 1.f / (1.f + __expf(-v));
    return v;
}

// Low 32 bits of a generic pointer to LDS == LDS byte offset (flat->LDS mapping
// truncates to addr[31:0], ISA 10.2).
__device__ __forceinline__ unsigned lds_off(const void* p) {
    return (unsigned)(uintptr_t)p;
}

// CDNA5 async DMA: global -> LDS, per-lane scatter, tracked by ASYNCcnt.
__device__ __forceinline__ void async_load_f32(const float* gp, float* lp) {
    asm volatile("global_load_async_to_lds_b32 %0, %1, off"
                 :: "v"(lds_off(lp)), "v"(gp) : "memory");
}
__device__ __forceinline__ void async_wait0() {
    asm volatile("s_wait_asynccnt 0x0" ::: "memory");
}

// -------------------------------------------------------------------------
// 1x1 conv as LDS-staged GEMM (V_WMMA_F32_16X16X4_F32).
// Block = (n, 32-pixel tile) x all Cout tiles; 8 waves cover coTiles*2
// 16x16 tiles (loop twice when Cout=100 -> 14 tiles; scalar-branch guard via
// readfirstlane so EXEC stays all-1s around WMMA).
// LDS layouts are K-innermost with +2 pad so every A/B fragment is one
// aligned ds_load_b64.  Staging uses global_load_async_to_lds_b32.
// -------------------------------------------------------------------------
#define KC    16
#define KP    18      // padded K stride (conflict-free, 8B-aligned pairs)
#define COPAD 112

__global__ __launch_bounds__(256)
void conv1x1_wmma_kernel(const float* __restrict__ x, const float* __restrict__ w,
                         const float* __restrict__ scale, const float* __restrict__ bias,
                         const float* __restrict__ mul, float* __restrict__ out,
                         int Cin, int Cout, int HW, int act)
{
    __shared__ float sX[32 * KP];      // [pixel][k]
    __shared__ float sW[COPAD * KP];   // [co][k]

    const int pixBlocks = HW >> 5;
    const int n    = blockIdx.x / pixBlocks;
    const int pix0 = (blockIdx.x % pixBlocks) << 5;

    const int tid   = threadIdx.x;
    const int waveU = __builtin_amdgcn_readfirstlane(tid >> 5);   // SGPR wave id
    const int lane  = tid & 31;
    const int half  = lane >> 4;
    const int l16   = lane & 15;

    const int coTiles = (Cout + 15) >> 4;
    const int T = coTiles * 2;          // 16x16 tiles in this block (<= 14)

    const float* xb = x + (size_t)n * Cin * HW + pix0;

    v8f acc[2];
    acc[0] = (v8f){0.f,0.f,0.f,0.f,0.f,0.f,0.f,0.f};
    acc[1] = (v8f){0.f,0.f,0.f,0.f,0.f,0.f,0.f,0.f};

    for (int kc = 0; kc < Cin; kc += KC) {
        __syncthreads();
        // async-stage X chunk (coalesced global rows of 32 floats)
        for (int i = tid; i < KC * 32; i += 256) {
            const int kk = i >> 5, pp = i & 31;
            async_load_f32(xb + (size_t)(kc + kk) * HW + pp, &sX[pp * KP + kk]);
        }
        // async-stage W chunk as [co][k]; zero-pad co >= Cout
        for (int i = tid; i < KC * COPAD; i += 256) {
            const int co = i / KC, kk = i % KC;
            float* dp = &sW[co * KP + kk];
            if (co < Cout) async_load_f32(w + (size_t)co * Cin + kc + kk, dp);
            else           *dp = 0.f;
        }
        async_wait0();
        __syncthreads();
#pragma unroll
        for (int u = 0; u < 2; ++u) {
            const int tl = waveU + u * 8;
            if (tl >= T) break;                  // scalar branch
            const int coT = tl >> 1, ph = tl & 1;
            const int m    = coT * 16 + l16;
            const int pcol = ph * 16 + l16;
#pragma unroll
            for (int k4 = 0; k4 < KC; k4 += 4) {
                const int ka = k4 + half * 2;
                const v2f a = *(const v2f*)&sW[m * KP + ka];
                const v2f b = *(const v2f*)&sX[pcol * KP + ka];
                acc[u] = __builtin_amdgcn_wmma_f32_16x16x4_f32(false, a, false, b,
                                                               (short)0, acc[u], false, false);
            }
        }
    }

#pragma unroll
    for (int u = 0; u < 2; ++u) {
        const int tl = waveU + u * 8;
        if (tl >= T) break;
        const int coT = tl >> 1, ph = tl & 1;
        const int p = pix0 + ph * 16 + l16;
        float* ob = out + (size_t)n * Cout * HW + p;
        const float* mb = mul ? (mul + (size_t)n * Cout * HW + p) : nullptr;
#pragma unroll
        for (int r = 0; r < 8; ++r) {
            const int mo = coT * 16 + half * 8 + r;
            if (mo < Cout) {
                float v = acc[u][r];
                if (scale) v *= scale[mo];
                v += bias[mo];
                v = act_apply(v, act);
                if (mb) v *= mb[(size_t)mo * HW];
                ob[(size_t)mo * HW] = v;
            }
        }
    }
}

// -------------------------------------------------------------------------
// 3x3 conv (stride 1, zero-pad 1), C=64, LDS-staged implicit GEMM.
// Block = (n, row h, 32-wide strip); 8 waves = 4 coTiles x 2 pixel halves.
// sIn layout [(row*34+col)][ci] (stride 66) and sWt [co][ci] (stride 66) so
// A/B fragments are aligned ds_load_b64, lane stride 66 dw = conflict-free.
// Halo borders pre-zeroed; optional xadd fused at staging (regular path),
// otherwise staging is async-to-LDS.
// -------------------------------------------------------------------------
__global__ __launch_bounds__(256)
void conv3x3_wmma_kernel(const float* __restrict__ x, const float* __restrict__ xadd,
                         const float* __restrict__ w, const float* __restrict__ scale,
                         const float* __restrict__ bias, float* __restrict__ out,
                         int H, int W, int act)
{
    __shared__ float sIn[3 * 34 * 66];   // [(row,col)][ci]
    __shared__ float sWt[64 * 66];       // [co][ci] for current tap

    const int HW = H * W;
    const int wBlocks = W >> 5;
    int bidx = blockIdx.x;
    const int w0 = (bidx % wBlocks) << 5;  bidx /= wBlocks;
    const int h  = bidx % H;               bidx /= H;
    const int n  = bidx;

    const int tid  = threadIdx.x;
    const int waveIn = tid >> 5;
    const int lane = tid & 31;
    const int half = lane >> 4;
    const int l16  = lane & 15;
    const int coT  = waveIn >> 1;        // 0..3
    const int ph   = waveIn & 1;         // 0..1
    const int m    = coT * 16 + l16;

    const float* xb  = x + (size_t)n * 64 * HW;
    const float* x2b = xadd ? (xadd + (size_t)n * 64 * HW) : nullptr;

    // stage halo: [(rr*34+cc)][c], zero-filled at borders, xadd fused
    for (int i = tid; i < 64 * 102; i += 256) {
        const int c  = i / 102;
        const int rc = i % 102;          // rr*34 + cc
        const int rr = rc / 34;
        const int cc = rc % 34;
        const int gh = h + rr - 1;
        const int gw = w0 + cc - 1;
        float* dp = &sIn[rc * 66 + c];
        if (gh >= 0 && gh < H && gw >= 0 && gw < W) {
            const size_t gi = (size_t)c * HW + (size_t)gh * W + gw;
            if (x2b) *dp = xb[gi] + x2b[gi];
            else     async_load_f32(xb + gi, dp);
        } else {
            *dp = 0.f;
        }
    }
    async_wait0();

    v8f acc = (v8f){0.f,0.f,0.f,0.f,0.f,0.f,0.f,0.f};
    for (int tap = 0; tap < 9; ++tap) {
        __syncthreads();                 // covers sIn staging on tap 0
        for (int i = tid; i < 64 * 64; i += 256) {
            const int co = i >> 6, ci = i & 63;
            async_load_f32(w + ((size_t)co * 64 + ci) * 9 + tap, &sWt[co * 66 + ci]);
        }
        async_wait0();
        __syncthreads();
        const int dy = tap / 3, dx = tap % 3;
        const float* sRow = &sIn[(dy * 34 + (ph * 16 + l16 + dx)) * 66];
#pragma unroll 4
        for (int k = 0; k < 64; k += 4) {
            const int ka = k + half * 2;
            const v2f a = *(const v2f*)&sWt[m * 66 + ka];
            const v2f b = *(const v2f*)&sRow[ka];
            acc = __builtin_amdgcn_wmma_f32_16x16x4_f32(false, a, false, b,
                                                        (short)0, acc, false, false);
        }
    }

    const int p = h * W + w0 + ph * 16 + l16;
    float* ob = out + (size_t)n * 64 * HW + p;
#pragma unroll
    for (int r = 0; r < 8; ++r) {
        const int mo = coT * 16 + half * 8 + r;
        float v = acc[r];
        if (scale) v *= scale[mo];
        v += bias[mo];
        ob[(size_t)mo * HW] = act_apply(v, act);
    }
}

// -------------------------------------------------------------------------
// MSDA branch convs: grouped 3x3, 16 groups, 4 in-ch -> 1 out-ch per group,
// dilations {1,2,4}, "same" zero pad.  Writes stacked [N,48,H,W].
// -------------------------------------------------------------------------
__global__ void msda_branches_kernel(const float* __restrict__ x,
                                     const float* __restrict__ lw, const float* __restrict__ lb,
                                     const float* __restrict__ gw, const float* __restrict__ gb,
                                     const float* __restrict__ ww, const float* __restrict__ wb,
                                     float* __restrict__ stacked, int N, int H, int W)
{
    const int HW = H * W;
    int idx = blockIdx.x * blockDim.x + threadIdx.x;
    if (idx >= N * 16 * HW) return;
    const int p  = idx % HW;
    int t        = idx / HW;
    const int co = t % 16;
    const int n  = t / 16;
    const int h  = p / W;
    const int wq = p % W;

    const float* xb = x + ((size_t)n * 64 + co * 4) * HW;
    const int dil[3] = {1, 2, 4};
    const float* wts[3] = {lw, gw, ww};
    const float* bs[3]  = {lb, gb, wb};

    for (int b = 0; b < 3; ++b) {
        const int d = dil[b];
        float sum = bs[b][co];
        const float* wt = wts[b] + (size_t)co * 4 * 9;
        for (int ci = 0; ci < 4; ++ci) {
            for (int dy = 0; dy < 3; ++dy) {
                const int h2 = h + (dy - 1) * d;
                if (h2 < 0 || h2 >= H) continue;
                for (int dx = 0; dx < 3; ++dx) {
                    const int w2 = wq + (dx - 1) * d;
                    if (w2 < 0 || w2 >= W) continue;
                    sum += wt[ci * 9 + dy * 3 + dx] * xb[(size_t)ci * HW + h2 * W + w2];
                }
            }
        }
        stacked[((size_t)n * 48 + b * 16 + co) * HW + p] = sum;
    }
}

// softmax over 100 channels of (ker * gate), written back into ker
__global__ void softmax100_kernel(float* __restrict__ ker, const float* __restrict__ gate,
                                  int N, int HW)
{
    int idx = blockIdx.x * blockDim.x + threadIdx.x;
    if (idx >= N * HW) return;
    const int p = idx % HW;
    const int n = idx / HW;
    const size_t base = (size_t)n * 100 * HW + p;

    float mx = -3.4e38f;
    for (int c = 0; c < 100; ++c)
        mx = fmaxf(mx, ker[base + (size_t)c * HW] * gate[base + (size_t)c * HW]);
    float sum = 0.f;
    for (int c = 0; c < 100; ++c) {
        float e = __expf(ker[base + (size_t)c * HW] * gate[base + (size_t)c * HW] - mx);
        ker[base + (size_t)c * HW] = e;
        sum += e;
    }
    const float inv = 1.f / sum;
    for (int c = 0; c < 100; ++c) ker[base + (size_t)c * HW] *= inv;
}

// CARAFE combine: out[n,c,2h+s,2w+t] = sum_{i,j} xpad[n,c,h+i-2,w+j-2]*k[n,((i*5+j)*2+s)*2+t,h,w]
// replicate padding on x.
__global__ void carafe_kernel(const float* __restrict__ x, const float* __restrict__ kk,
                              float* __restrict__ out, int N, int C, int H, int W)
{
    const int Ho = 2 * H, Wo = 2 * W;
    int idx = blockIdx.x * blockDim.x + threadIdx.x;
    if (idx >= N * C * Ho * Wo) return;
    const int wo = idx % Wo;
    int t        = idx / Wo;
    const int ho = t % Ho;  t /= Ho;
    const int c  = t % C;
    const int n  = t / C;

    const int h = ho >> 1, s = ho & 1;
    const int w0 = wo >> 1, t0 = wo & 1;
    const float* xb = x + ((size_t)n * C + c) * H * W;
    const float* kb = kk + (size_t)n * 100 * H * W + (size_t)h * W + w0;

    float sum = 0.f;
#pragma unroll
    for (int i = 0; i < 5; ++i) {
        const int hi = min(max(h + i - 2, 0), H - 1);
#pragma unroll
        for (int j = 0; j < 5; ++j) {
            const int wj = min(max(w0 + j - 2, 0), W - 1);
            const int kc = ((i * 5 + j) * 2 + s) * 2 + t0;
            sum += xb[(size_t)hi * W + wj] * kb[(size_t)kc * H * W];
        }
    }
    out[((size_t)(n * C + c) * Ho + ho) * Wo + wo] = sum;
}

// -------------------------------------------------------------------------
extern "C" void kernel_launch(void* const* d_in, const int* in_sizes, int n_in,
                              void* d_out, int out_size, void* d_ws, size_t ws_size,
                              hipStream_t stream)
{
    const float* IN[64];
    for (int i = 0; i < n_in && i < 64; ++i) IN[i] = (const float*)d_in[i];
    float* ws  = (float*)d_ws;
    float* out = (float*)d_out;
    const int N = 8;

    // Input index map (setup_inputs dict insertion order):
    // 0 p3, 1 p4, 2 p5
    // 3..5 lat3{w,s,b}  6..8 lat4  9..11 lat5
    // 12..14 sm3{w,s,b} 15..17 sm4
    // 18..25 msda3{lw,lb,gw,gb,ww,wb,fw,fb}  26..33 msda4  34..41 msda5
    // 42..50 sau54{enc_w,enc_s,enc_b,kw,kb,g1w,g1b,g2w,g2b}  51..59 sau43

    size_t o = 0;
    float* lat3  = ws + o; o += (size_t)N * 64 * 128 * 128;
    float* lat4  = ws + o; o += (size_t)N * 64 * 64 * 64;
    float* lat5  = ws + o; o += (size_t)N * 64 * 32 * 32;
    float* stk   = ws + o; o += (size_t)N * 48 * 128 * 128;   // shared MSDA stacked
    float* encB  = ws + o; o += (size_t)N * 64 * 64 * 64;     // shared SAU enc (max size)
    float* gB    = ws + o; o += (size_t)N * 64 * 64 * 64;
    float* kerB  = ws + o; o += (size_t)N * 100 * 64 * 64;
    float* gateB = ws + o; o += (size_t)N * 100 * 64 * 64;
    float* upB   = ws + o; o += (size_t)N * 64 * 128 * 128;
    float* fused = ws + o; o += (size_t)N * 64 * 128 * 128;
    (void)ws_size;

    float* out_p3 = out;
    float* out_p4 = out + (size_t)N * 64 * 128 * 128;
    float* out_p5 = out_p4 + (size_t)N * 64 * 64 * 64;

    auto c1x1 = [&](const float* x, const float* w, const float* s, const float* b,
                    const float* mul, float* dst, int Cin, int Cout, int HW, int act) {
        int blocks = N * (HW / 32);
        conv1x1_wmma_kernel<<<blocks, 256, 0, stream>>>(x, w, s, b, mul, dst,
                                                        Cin, Cout, HW, act);
    };
    auto c3x3 = [&](const float* x, const float* xadd, const float* w, const float* s,
                    const float* b, float* dst, int H, int W, int act) {
        int blocks = N * H * (W / 32);
        conv3x3_wmma_kernel<<<blocks, 256, 0, stream>>>(x, xadd, w, s, b, dst, H, W, act);
    };
    auto msda = [&](const float* x, const float* const* mp, float* dst, int H, int W) {
        int HW = H * W;
        int thr = N * 16 * HW;
        msda_branches_kernel<<<(thr + 255) / 256, 256, 0, stream>>>(
            x, mp[0], mp[1], mp[2], mp[3], mp[4], mp[5], stk, N, H, W);
        c1x1(stk, mp[6], nullptr, mp[7], /*mul=*/x, dst, 48, 64, HW, ACT_SIGMOID);
    };
    auto sau = [&](const float* x, const float* const* sp, int H, int W) {
        int HW = H * W;
        c3x3(x, nullptr, sp[0], sp[1], sp[2], encB, H, W, ACT_RELU);
        c1x1(encB, sp[3], nullptr, sp[4], nullptr, kerB, 64, 100, HW, ACT_NONE);
        c3x3(encB, nullptr, sp[5], nullptr, sp[6], gB, H, W, ACT_RELU);
        c1x1(gB, sp[7], nullptr, sp[8], nullptr, gateB, 64, 100, HW, ACT_SIGMOID);
        int tp = N * HW;
        softmax100_kernel<<<(tp + 255) / 256, 256, 0, stream>>>(kerB, gateB, N, HW);
        int to = N * 64 * 2 * H * 2 * W;
        carafe_kernel<<<(to + 255) / 256, 256, 0, stream>>>(x, kerB, upB, N, 64, H, W);
    };

    const float* msda3p[8] = {IN[18],IN[19],IN[20],IN[21],IN[22],IN[23],IN[24],IN[25]};
    const float* msda4p[8] = {IN[26],IN[27],IN[28],IN[29],IN[30],IN[31],IN[32],IN[33]};
    const float* msda5p[8] = {IN[34],IN[35],IN[36],IN[37],IN[38],IN[39],IN[40],IN[41]};
    const float* sau54p[9] = {IN[42],IN[43],IN[44],IN[45],IN[46],IN[47],IN[48],IN[49],IN[50]};
    const float* sau43p[9] = {IN[51],IN[52],IN[53],IN[54],IN[55],IN[56],IN[57],IN[58],IN[59]};

    // lateral 1x1 conv + BN + SiLU
    c1x1(IN[2], IN[9],  IN[10], IN[11], nullptr, lat5, 512, 64, 32 * 32,   ACT_SILU);
    c1x1(IN[1], IN[6],  IN[7],  IN[8],  nullptr, lat4, 256, 64, 64 * 64,   ACT_SILU);
    c1x1(IN[0], IN[3],  IN[4],  IN[5],  nullptr, lat3, 128, 64, 128 * 128, ACT_SILU);

    // P5 branch
    msda(lat5, msda5p, out_p5, 32, 32);
    sau(out_p5, sau54p, 32, 32);                       // -> upB (N,64,64,64)

    // P4 branch
    c3x3(lat4, upB, IN[15], IN[16], IN[17], fused, 64, 64, ACT_SILU);
    msda(fused, msda4p, out_p4, 64, 64);
    sau(out_p4, sau43p, 64, 64);                       // -> upB (N,64,128,128)

    // P3 branch
    c3x3(lat3, upB, IN[12], IN[13], IN[14], fused, 128, 128, ACT_SILU);
    msda(fused, msda3p, out_p3, 128, 128);
}